// FocalLoss_47167330845121
// MI455X (gfx1250) — compile-verified
//
#include <hip/hip_runtime.h>
#include <cstdint>

#define DEV_INLINE __device__ __forceinline__

// Problem geometry (reference: N=4096 rows, C=32000 classes).
constexpr int   kC      = 32000;
constexpr int   kBlock  = 256;                         // 8 wave32
constexpr int   kChunk  = kBlock * 4;                  // 1024 floats = 4 KB per chunk
constexpr int   kNChunk = (kC + kChunk - 1) / kChunk;  // 32 chunks (last partial: 256 floats)
constexpr int   kDepth  = 4;                           // LDS ring depth (issue-ahead = 3)
constexpr float kLog2e  = 1.44269504088896340736f;
constexpr float kLn2    = 0.69314718055994530942f;

// 16-byte vector type matching the async-LDS builtin's pointee type.
typedef int vint4_t __attribute__((vector_size(4 * sizeof(int))));
typedef __attribute__((address_space(1))) vint4_t* global_v4i_p;
typedef __attribute__((address_space(3))) vint4_t* lds_v4i_p;

// ---- CDNA5 async global->LDS copy (ASYNCcnt path) ----------------------------
DEV_INLINE void async_copy_b128(const float* g, float* l) {
#if __has_builtin(__builtin_amdgcn_global_load_async_to_lds_b128)
  __builtin_amdgcn_global_load_async_to_lds_b128(
      (global_v4i_p)g, (lds_v4i_p)l, 0, 0);
#else
  unsigned lds_off =
      (unsigned)(unsigned long long)(__attribute__((address_space(3))) void*)l;
  asm volatile("global_load_async_to_lds_b128 %0, %1, off"
               :: "v"(lds_off), "v"((unsigned long long)g)
               : "memory");
#endif
}

template <int N> DEV_INLINE void wait_async() {
#if __has_builtin(__builtin_amdgcn_s_wait_asynccnt)
  __builtin_amdgcn_s_wait_asynccnt(N);
#else
  asm volatile("s_wait_asynccnt %0" :: "i"(N) : "memory");
#endif
}

// ---- fast transcendental helpers (v_exp_f32 / v_log_f32 are base-2) ----------
DEV_INLINE float fast_exp2(float x) {
#if __has_builtin(__builtin_amdgcn_exp2f)
  return __builtin_amdgcn_exp2f(x);
#else
  return exp2f(x);
#endif
}
DEV_INLINE float fast_log2(float x) {
#if __has_builtin(__builtin_amdgcn_logf)
  return __builtin_amdgcn_logf(x);
#else
  return log2f(x);
#endif
}

// Online softmax update in base-2: m = running max of y, s = sum 2^(y-m).
// Common path costs exactly one v_exp_f32.
DEV_INLINE void online_update(float y, int idx, float& m, float& s, float& bv, int& bi) {
  if (y > bv) { bv = y; bi = idx; }
  if (y <= m) {
    s += fast_exp2(y - m);
  } else {
    s = s * fast_exp2(m - y) + 1.0f;  // first call: m=-inf -> exp2(-inf)=0, s=1
    m = y;
  }
}

// ---- Kernel 1: one block per row, async-LDS pipelined online softmax ---------
__global__ __launch_bounds__(kBlock) void focal_row_kernel(
    const float* __restrict__ input,   // [N, C]
    const int*   __restrict__ target,  // [N]
    const float* __restrict__ weight,  // [C]
    const float* __restrict__ gamma,   // [1]
    float*       __restrict__ row_loss /* [N] */) {
  __shared__ __align__(16) float buf[kDepth][kChunk];
  __shared__ float rm[kBlock / 32], rs[kBlock / 32], rbv[kBlock / 32];
  __shared__ int   rbi[kBlock / 32];

  const int tid = threadIdx.x;
  const int r   = blockIdx.x;
  const float* row = input + (size_t)r * kC;

  // Prologue: issue chunks 0..kDepth-2 (issue-ahead depth kDepth-1).
#pragma unroll
  for (int j = 0; j < kDepth - 1; ++j) {
    const int col = j * kChunk + tid * 4;
    if (j < kNChunk && col < kC)
      async_copy_b128(row + col, &buf[j % kDepth][tid * 4]);
  }

  float m = -__builtin_inff();
  float s = 0.0f;
  float bv = -__builtin_inff();
  int   bi = 0;

  for (int k = 0; k < kNChunk; ++k) {
    // Chunk k arrived  <=>  outstanding asyncs <= min(kDepth-2, chunks issued past k).
    int ahead = kNChunk - 1 - k;
    if (ahead > kDepth - 2) ahead = kDepth - 2;
    switch (ahead) {
      case 0:  wait_async<0>(); break;
      case 1:  wait_async<1>(); break;
      default: wait_async<kDepth - 2>(); break;
    }
    __syncthreads();  // chunk k globally visible; all waves done reading buf[(k-1)%D]

    // Refill the ring slot freed last iteration with chunk k+kDepth-1.
    const int j = k + kDepth - 1;
    if (j < kNChunk) {
      const int jcol = j * kChunk + tid * 4;
      if (jcol < kC)
        async_copy_b128(row + jcol, &buf[j % kDepth][tid * 4]);
    }

    // Consume chunk k from LDS (ds_load_b128), overlapped with the load above.
    const int col = k * kChunk + tid * 4;
    if (col < kC) {
      const float4 v = *reinterpret_cast<const float4*>(&buf[k % kDepth][tid * 4]);
      online_update(v.x * kLog2e, col + 0, m, s, bv, bi);
      online_update(v.y * kLog2e, col + 1, m, s, bv, bi);
      online_update(v.z * kLog2e, col + 2, m, s, bv, bi);
      online_update(v.w * kLog2e, col + 3, m, s, bv, bi);
    }
  }

  // ---- wave32 reduction (combine (m,s) pairs + argmax, first-index tiebreak) --
#pragma unroll
  for (int off = 16; off > 0; off >>= 1) {
    const float m2  = __shfl_xor(m,  off, 32);
    const float s2  = __shfl_xor(s,  off, 32);
    const float bv2 = __shfl_xor(bv, off, 32);
    const int   bi2 = __shfl_xor(bi, off, 32);
    const float M = fmaxf(m, m2);
    s = s * fast_exp2(m - M) + s2 * fast_exp2(m2 - M);
    m = M;
    if (bv2 > bv || (bv2 == bv && bi2 < bi)) { bv = bv2; bi = bi2; }
  }

  const int wave = tid >> 5, lane = tid & 31;
  if (lane == 0) { rm[wave] = m; rs[wave] = s; rbv[wave] = bv; rbi[wave] = bi; }
  __syncthreads();

  if (tid == 0) {
    float M = rm[0], S = rs[0], BV = rbv[0];
    int   BI = rbi[0];
#pragma unroll
    for (int i = 1; i < kBlock / 32; ++i) {
      const float M2 = fmaxf(M, rm[i]);
      S = S * fast_exp2(M - M2) + rs[i] * fast_exp2(rm[i] - M2);
      M = M2;
      if (rbv[i] > BV || (rbv[i] == BV && rbi[i] < BI)) { BV = rbv[i]; BI = rbi[i]; }
    }
    // Final scalar math: logp_t, p_t, weight[argmax], (1-p_t)^gamma.
    const int   tgt    = target[r];
    const float yt     = row[tgt] * kLog2e;
    const float log2pt = yt - M - fast_log2(S);   // log2 p_t  (S >= 1)
    const float logpt  = log2pt * kLn2;           // ln p_t
    const float pt     = fast_exp2(log2pt);
    const float w      = weight[BI];
    const float g      = gamma[0];
    const float om     = fmaxf(1.0f - pt, 0.0f);
    const float fw     = (om > 0.0f) ? fast_exp2(g * fast_log2(om)) : 0.0f;
    row_loss[r] = w * fw * (-logpt);
  }
}

// ---- Kernel 2: deterministic 4096 -> 1 reduction -----------------------------
__global__ __launch_bounds__(256) void reduce_sum_kernel(
    const float* __restrict__ in, float* __restrict__ out, int n) {
  __shared__ float sm[256];
  float acc = 0.0f;
  for (int i = threadIdx.x; i < n; i += 256) acc += in[i];
  sm[threadIdx.x] = acc;
  __syncthreads();
#pragma unroll
  for (int stp = 128; stp > 0; stp >>= 1) {
    if ((int)threadIdx.x < stp) sm[threadIdx.x] += sm[threadIdx.x + stp];
    __syncthreads();
  }
  if (threadIdx.x == 0) out[0] = sm[0];
}

extern "C" void kernel_launch(void* const* d_in, const int* in_sizes, int n_in,
                              void* d_out, int out_size, void* d_ws, size_t ws_size,
                              hipStream_t stream) {
  const float* input  = (const float*)d_in[0];   // [N, C] f32
  const int*   target = (const int*)d_in[1];     // [N] int
  const float* weight = (const float*)d_in[2];   // [C] f32
  const float* gamma  = (const float*)d_in[3];   // [1] f32
  float* partials = (float*)d_ws;                // N floats of scratch
  float* out      = (float*)d_out;               // scalar

  const int N = in_sizes[1];                     // rows == len(target)

  focal_row_kernel<<<dim3(N), dim3(kBlock), 0, stream>>>(input, target, weight,
                                                         gamma, partials);
  reduce_sum_kernel<<<dim3(1), dim3(256), 0, stream>>>(partials, out, N);
}